// MindieCifTimestamp_71244917506096
// MI455X (gfx1250) — compile-verified
//
#include <hip/hip_runtime.h>
#include <cstdint>

// CIF scan, threshold = 1.0.
// fires[b,t] = P_t - floor(P_{t-1}),  integrate_new[b] = P_{T-1} - floor(P_{T-1})
// where P_t = integrate[b] + sum_{i<=t} alpha[b,i]   (valid since alpha,integrate in [0,1)).

constexpr int LANES = 32;            // wave32
constexpr int VEC   = 4;             // floats per lane per scan step
constexpr int STEPF = LANES * VEC;   // 128 floats per wave scan step
constexpr int CHUNK = 1024;          // floats per LDS buffer per wave (4 KB)
constexpr int ITERS = CHUNK / STEPF; // 8 scan steps per chunk
constexpr int WAVES = 8;             // waves per block -> 64 KB LDS per block

__global__ __launch_bounds__(WAVES * LANES)
void cif_scan_kernel(const float* __restrict__ alphas,
                     const float* __restrict__ integrate,
                     float* __restrict__ fires,
                     float* __restrict__ integ_out,
                     int B, int T)
{
    __shared__ float lds[WAVES * 2 * CHUNK];

    const int lane = threadIdx.x & (LANES - 1);
    const int wave = threadIdx.x >> 5;
    const int row  = blockIdx.x * WAVES + wave;
    if (row >= B) return;

    const int nchunk = T / CHUNK;

    const float* __restrict__ rowIn  = alphas + (size_t)row * T;
    float*       __restrict__ rowOut = fires  + (size_t)row * T;

    float* buf[2] = { &lds[(wave * 2 + 0) * CHUNK], &lds[(wave * 2 + 1) * CHUNK] };
    // Low 32 bits of a flat LDS pointer == byte offset within the workgroup LDS
    // allocation.
    uint32_t ldsOff[2] = { (uint32_t)(uintptr_t)buf[0], (uint32_t)(uintptr_t)buf[1] };

    // Issue 8 async b128 loads covering one CHUNK (32 lanes x 16 B = 512 B each).
    auto issue = [&](int c, uint32_t lbase) {
        unsigned long long g = (unsigned long long)(uintptr_t)(rowIn + (size_t)c * CHUNK)
                             + (unsigned long long)lane * 16ull;
        uint32_t l = lbase + (uint32_t)lane * 16u;
        #pragma unroll
        for (int i = 0; i < (CHUNK * 4) / (LANES * 16); ++i) {
            asm volatile("global_load_async_to_lds_b128 %0, %1, off"
                         :: "v"(l), "v"(g) : "memory");
            g += (unsigned long long)(LANES * 16);
            l += (uint32_t)(LANES * 16);
        }
    };

    float R = integrate[row];   // running prefix total P (uniform across the wave)

    issue(0, ldsOff[0]);

    for (int c = 0; c < nchunk; ++c) {
        const float* cur = buf[c & 1];
        if (c + 1 < nchunk) {
            issue(c + 1, ldsOff[(c + 1) & 1]);
            // async loads complete in order: <=8 outstanding => chunk c has landed
            asm volatile("s_wait_asynccnt 8" ::: "memory");
        } else {
            asm volatile("s_wait_asynccnt 0" ::: "memory");
        }

        #pragma unroll
        for (int it = 0; it < ITERS; ++it) {
            const float4 a = *(const float4*)(cur + it * STEPF + lane * VEC);

            // local inclusive prefix within the lane's 4 elements
            float p0 = a.x;
            float p1 = p0 + a.y;
            float p2 = p1 + a.z;
            float p3 = p2 + a.w;

            // wave-wide inclusive scan of lane totals (5 shuffle steps, wave32)
            float s = p3;
            #pragma unroll
            for (int d = 1; d < LANES; d <<= 1) {
                float t = __shfl_up(s, d, LANES);
                if (lane >= d) s += t;
            }
            float excl = __shfl_up(s, 1, LANES);
            if (lane == 0) excl = 0.0f;

            float base = R + excl;         // P just before this lane's first element
            float P0 = base + p0;
            float P1 = base + p1;
            float P2 = base + p2;
            float P3 = base + p3;

            float4 f;
            f.x = P0 - floorf(base);
            f.y = P1 - floorf(P0);
            f.z = P2 - floorf(P1);
            f.w = P3 - floorf(P2);
            *(float4*)(rowOut + (size_t)c * CHUNK + it * STEPF + lane * VEC) = f;

            // carry: new running total = old R + 128-element block sum (lane 31)
            R = __shfl(R + s, LANES - 1, LANES);
        }
        // ensure all LDS reads of this buffer retired before it is re-targeted
        asm volatile("s_wait_dscnt 0" ::: "memory");
    }

    if (lane == 0) integ_out[row] = R - floorf(R);
}

extern "C" void kernel_launch(void* const* d_in, const int* in_sizes, int n_in,
                              void* d_out, int out_size, void* d_ws, size_t ws_size,
                              hipStream_t stream)
{
    const float* alphas    = (const float*)d_in[0];   // [B, T] f32
    const float* integrate = (const float*)d_in[1];   // [B]    f32
    const int B = in_sizes[1];
    const int T = in_sizes[0] / B;

    float* fires     = (float*)d_out;                        // [B, T]
    float* integ_out = (float*)d_out + (size_t)B * T;        // [B]

    const int blocks = (B + WAVES - 1) / WAVES;
    hipLaunchKernelGGL(cif_scan_kernel, dim3(blocks), dim3(WAVES * LANES), 0, stream,
                       alphas, integrate, fires, integ_out, B, T);
}